// GraphSAGE_3530463117553
// MI455X (gfx1250) — compile-verified
//
#include <hip/hip_runtime.h>

typedef float v2f __attribute__((ext_vector_type(2)));
typedef float v8f __attribute__((ext_vector_type(8)));

#define N_NODES 100000
#define F_IN    128
#define H_DIM   128
#define C_DIM   40
#define LDS_STRIDE 132   // 128 + 4 pad: bank = (4*m + k) % 64 -> conflict-free WMMA A reads

// ---------------------------------------------------------------------------
// Zero the aggregation buffer (ws is poisoned, and we must be replay-safe).
// ---------------------------------------------------------------------------
__global__ __launch_bounds__(256) void zero_f4(float4* __restrict__ p, int n4) {
    int i = blockIdx.x * 256 + threadIdx.x;
    if (i < n4) p[i] = make_float4(0.f, 0.f, 0.f, 0.f);
}

// ---------------------------------------------------------------------------
// Edge scatter-add: agg[dst] += feat[src].  32 lanes per edge, float4 gather,
// 4x global_atomic_add_f32 per lane.  agg (51MB) is L2-resident on MI455X.
// ---------------------------------------------------------------------------
__global__ __launch_bounds__(256) void scatter_add(const float* __restrict__ feat,
                                                   const long long* __restrict__ ei,
                                                   float* __restrict__ agg, int E) {
    int tid  = blockIdx.x * 256 + threadIdx.x;
    int e    = tid >> 5;
    int lane = tid & 31;
    if (e >= E) return;
    long long s = ei[e];                 // edge_index[0][e]
    long long d = ei[(long long)E + e];  // edge_index[1][e]
    const float4 v = *(const float4*)(feat + s * (long long)F_IN + lane * 4);
    float* base = agg + d * (long long)F_IN + lane * 4;
    atomicAdd(base + 0, v.x);
    atomicAdd(base + 1, v.y);
    atomicAdd(base + 2, v.z);
    atomicAdd(base + 3, v.w);
}

// ---------------------------------------------------------------------------
// Layer 1: h = relu(agg @ W1_rel^T + b1 + x @ W1_root^T)
// Block = 16 rows x 128 cols; 8 waves, each wave one 16x16 tile via
// 2x32 V_WMMA_F32_16X16X4_F32 (full fp32 precision).
// ---------------------------------------------------------------------------
__global__ __launch_bounds__(256) void gemm1_relu(const float* __restrict__ agg,
                                                  const float* __restrict__ x,
                                                  const float* __restrict__ Wrel,
                                                  const float* __restrict__ bias,
                                                  const float* __restrict__ Wroot,
                                                  float* __restrict__ h) {
    __shared__ float As[16 * LDS_STRIDE];
    __shared__ float Xs[16 * LDS_STRIDE];
    const int row0 = blockIdx.x * 16;
    const int tid  = threadIdx.x;

    // Stage 16x128 A-tiles (agg and x) into LDS: 512 float4 each, 2 per thread.
    for (int c = tid; c < 512; c += 256) {
        int r = c >> 5, col4 = (c & 31) << 2;
        *(float4*)&As[r * LDS_STRIDE + col4] =
            *(const float4*)(agg + (long long)(row0 + r) * F_IN + col4);
        *(float4*)&Xs[r * LDS_STRIDE + col4] =
            *(const float4*)(x + (long long)(row0 + r) * F_IN + col4);
    }
    __syncthreads();

    const int wave  = tid >> 5;
    const int lane  = tid & 31;
    const int col0  = wave * 16;
    const int mn    = lane & 15;          // M for A, N for B/C
    const int khalf = (lane >> 4) * 2;    // 0 or 2: which K-pair this half-wave owns

    v8f c;
    const float bv = bias[col0 + mn];     // C/D: every VGPR holds column N=lane%16
    #pragma unroll
    for (int i = 0; i < 8; i++) c[i] = bv;

    #pragma unroll 4
    for (int k0 = 0; k0 < F_IN; k0 += 4) {
        const int kk = k0 + khalf;
        v2f a  = *(const v2f*)&As[mn * LDS_STRIDE + kk];
        v2f b  = *(const v2f*)(Wrel + (long long)(col0 + mn) * F_IN + kk);
        c = __builtin_amdgcn_wmma_f32_16x16x4_f32(false, a, false, b, (short)0, c, false, false);
        v2f a2 = *(const v2f*)&Xs[mn * LDS_STRIDE + kk];
        v2f b2 = *(const v2f*)(Wroot + (long long)(col0 + mn) * F_IN + kk);
        c = __builtin_amdgcn_wmma_f32_16x16x4_f32(false, a2, false, b2, (short)0, c, false, false);
    }

    const int rbase = (lane >> 4) * 8;    // D: M = vgpr + 8*(lane/16)
    #pragma unroll
    for (int r = 0; r < 8; r++) {
        float v = c[r] > 0.f ? c[r] : 0.f;
        h[(long long)(row0 + rbase + r) * H_DIM + col0 + mn] = v;
    }
}

// ---------------------------------------------------------------------------
// Layer 2 + fused log_softmax: out = log_softmax(agg@W2_rel^T + b2 + h@W2_root^T)
// Block = 16 rows x 48 cols (40 valid); 3 waves.  Softmax done in-block.
// ---------------------------------------------------------------------------
__global__ __launch_bounds__(96) void gemm2_lsm(const float* __restrict__ agg,
                                                const float* __restrict__ h,
                                                const float* __restrict__ Wrel,
                                                const float* __restrict__ bias,
                                                const float* __restrict__ Wroot,
                                                float* __restrict__ out) {
    __shared__ float As[16 * LDS_STRIDE];
    __shared__ float Hs[16 * LDS_STRIDE];
    __shared__ float Ot[16 * 49];
    __shared__ float rowMax[16], rowLse[16];
    const int row0 = blockIdx.x * 16;
    const int tid  = threadIdx.x;

    for (int c = tid; c < 512; c += 96) {
        int r = c >> 5, col4 = (c & 31) << 2;
        *(float4*)&As[r * LDS_STRIDE + col4] =
            *(const float4*)(agg + (long long)(row0 + r) * H_DIM + col4);
        *(float4*)&Hs[r * LDS_STRIDE + col4] =
            *(const float4*)(h + (long long)(row0 + r) * H_DIM + col4);
    }
    __syncthreads();

    const int wave  = tid >> 5;           // 0..2
    const int lane  = tid & 31;
    const int col0  = wave * 16;
    const int mn    = lane & 15;
    const int khalf = (lane >> 4) * 2;
    const int col   = col0 + mn;
    const bool valid = (col < C_DIM);
    const int ccol  = valid ? col : 0;    // clamp to avoid OOB weight reads

    v8f c;
    const float bv = valid ? bias[col] : 0.f;
    #pragma unroll
    for (int i = 0; i < 8; i++) c[i] = bv;

    #pragma unroll 4
    for (int k0 = 0; k0 < H_DIM; k0 += 4) {
        const int kk = k0 + khalf;
        v2f a  = *(const v2f*)&As[mn * LDS_STRIDE + kk];
        v2f b  = *(const v2f*)(Wrel + (long long)ccol * H_DIM + kk);
        if (!valid) { b.x = 0.f; b.y = 0.f; }
        c = __builtin_amdgcn_wmma_f32_16x16x4_f32(false, a, false, b, (short)0, c, false, false);
        v2f a2 = *(const v2f*)&Hs[mn * LDS_STRIDE + kk];
        v2f b2 = *(const v2f*)(Wroot + (long long)ccol * H_DIM + kk);
        if (!valid) { b2.x = 0.f; b2.y = 0.f; }
        c = __builtin_amdgcn_wmma_f32_16x16x4_f32(false, a2, false, b2, (short)0, c, false, false);
    }

    const int rbase = (lane >> 4) * 8;
    #pragma unroll
    for (int r = 0; r < 8; r++) Ot[(rbase + r) * 49 + col] = c[r];
    __syncthreads();

    if (tid < 16) {
        float m = -3.402823466e38f;
        for (int cc = 0; cc < C_DIM; cc++) m = fmaxf(m, Ot[tid * 49 + cc]);
        float s = 0.f;
        for (int cc = 0; cc < C_DIM; cc++) s += __expf(Ot[tid * 49 + cc] - m);
        rowMax[tid] = m;
        rowLse[tid] = __logf(s);
    }
    __syncthreads();

    for (int idx = tid; idx < 16 * C_DIM; idx += 96) {
        int r = idx / C_DIM, cc = idx - r * C_DIM;
        out[(long long)(row0 + r) * C_DIM + cc] = Ot[r * 49 + cc] - rowMax[r] - rowLse[r];
    }
}

// ---------------------------------------------------------------------------
extern "C" void kernel_launch(void* const* d_in, const int* in_sizes, int n_in,
                              void* d_out, int out_size, void* d_ws, size_t ws_size,
                              hipStream_t stream) {
    const float*     x       = (const float*)d_in[0];
    const long long* ei      = (const long long*)d_in[1];
    const float*     W1_rel  = (const float*)d_in[2];
    const float*     b1      = (const float*)d_in[3];
    const float*     W1_root = (const float*)d_in[4];
    const float*     W2_rel  = (const float*)d_in[5];
    const float*     b2      = (const float*)d_in[6];
    const float*     W2_root = (const float*)d_in[7];
    float* out = (float*)d_out;

    float* agg = (float*)d_ws;                         // N*128 f32 (51.2 MB)
    float* h   = agg + (size_t)N_NODES * H_DIM;        // N*128 f32 (51.2 MB)

    const int E  = in_sizes[1] / 2;
    const int n4 = N_NODES * F_IN / 4;
    const int zeroGrid    = (n4 + 255) / 256;
    const int scatterGrid = (E * 32 + 255) / 256;      // 32 lanes per edge
    const int gemmGrid    = N_NODES / 16;              // 100000 = 6250*16 exactly

    // Layer 1
    zero_f4<<<zeroGrid, 256, 0, stream>>>((float4*)agg, n4);
    scatter_add<<<scatterGrid, 256, 0, stream>>>(x, ei, agg, E);
    gemm1_relu<<<gemmGrid, 256, 0, stream>>>(agg, x, W1_rel, b1, W1_root, h);

    // Layer 2
    zero_f4<<<zeroGrid, 256, 0, stream>>>((float4*)agg, n4);
    scatter_add<<<scatterGrid, 256, 0, stream>>>(h, ei, agg, E);
    gemm2_lsm<<<gemmGrid, 96, 0, stream>>>(agg, h, W2_rel, b2, W2_root, out);
}